// AttentionModule_38585986188023
// MI455X (gfx1250) — compile-verified
//
#include <hip/hip_runtime.h>

// ---------------------------------------------------------------------------
// Bahdanau-style attention scores, algebraically folded.
//
//   S=2048, B=32, H=512.  out[b,0,s] = softmax_s( w2 . enc[s,b,:] )
//   where w2[f] = sum_h v[h]*W[h,512+f]   (the hidden- and bias-dependent
//   score terms are constant in s and cancel exactly under the softmax).
//
// Cost model (MI455X): enc is 2048*32*1024*4 B = 256 MB streamed once
// -> ~11 us floor at 23.3 TB/s.  Everything else (<3 MB) is noise.
// The GEMV reduction runs on V_WMMA_F32_16X16X4_F32 (exact fp32):
// one wave owns 16 rows; B-matrix = w2 chunk broadcast across columns.
// ---------------------------------------------------------------------------

typedef __attribute__((ext_vector_type(2))) float v2f;
typedef __attribute__((ext_vector_type(8))) float v8f;

#define S_LEN 2048
#define B_DIM 32
#define H_DIM 512
#define F2    1024               // 2H : reduction length of the GEMV
#define ROWS  (S_LEN * B_DIM)    // 65536 flattened (s,b) rows

// --- Kernel 1: fold v into W:  w2[f] = sum_h v[h] * W[h, 512+f], f<1024 ----
__global__ void __launch_bounds__(256)
fold_w_kernel(const float* __restrict__ W, const float* __restrict__ v,
              float* __restrict__ w2) {
    int f = blockIdx.x * 256 + threadIdx.x;
    if (f >= F2) return;
    const float* col = W + H_DIM + f;        // W row stride = 3H = 1536
    float acc = 0.f;
#pragma unroll 8
    for (int h = 0; h < H_DIM; ++h)
        acc += v[h] * col[h * 1536];         // coalesced across f
    w2[f] = acc;
}

// --- Kernel 2: scores[s,b] = enc[s,b,:] . w2  via V_WMMA_F32_16X16X4_F32 ---
// Wave tile = 16 rows. A (16x4 fp32, ISA layout): lanes 0-15 hold K=0,1 of
// row M=lane; lanes 16-31 hold K=2,3 -> one float2 per lane per step.
// B (4x16): value depends only on K -> broadcast w2 chunk from LDS with the
// mirrored K layout.  All 16 D columns are identical; lanes 0/16 store.
__global__ void __launch_bounds__(256)
score_wmma_kernel(const float* __restrict__ enc,
                  const float* __restrict__ w2,
                  float* __restrict__ scores /* [B][S] */) {
    __shared__ float w2s[F2];                          // 4 KB
    for (int i = threadIdx.x; i < F2; i += 256) w2s[i] = w2[i];
    __syncthreads();

    const int wave     = threadIdx.x >> 5;
    const int lane     = threadIdx.x & 31;
    const int tile     = blockIdx.x * 8 + wave;        // 8 waves / block
    const int row_base = tile * 16;                    // 16 rows / wave
    const int m        = lane & 15;                    // row within tile
    const int hi       = lane >> 4;                    // K-half select

    const float* rowp = enc + (size_t)(row_base + m) * F2 + hi * 2;
    const float* w2p  = w2s + hi * 2;

    v8f acc = {0.f, 0.f, 0.f, 0.f, 0.f, 0.f, 0.f, 0.f};

#pragma unroll 4
    for (int k = 0; k < F2; k += 4) {
        if ((k & 31) == 0)                              // every 8 steps:
            __builtin_prefetch(rowp + k + 512, 0, 3);   // WGP-scope, 2 KB ahead
        v2f a = *(const v2f*)(rowp + k);                // global_load_b64
        v2f b = *(const v2f*)(w2p  + k);                // ds broadcast
        acc = __builtin_amdgcn_wmma_f32_16x16x4_f32(
                  /*neg_a=*/false, a, /*neg_b=*/false, b,
                  /*c_mod=*/(short)0, acc,
                  /*reuse_a=*/false, /*reuse_b=*/false);
    }

    // D layout: VGPR j -> M=j (lanes 0-15) / M=j+8 (lanes 16-31); columns
    // identical. Lane 0 stores rows 0..7, lane 16 stores rows 8..15.
    if (m == 0) {
        int rb = row_base + hi * 8;
#pragma unroll
        for (int j = 0; j < 8; ++j) {
            int r = rb + j;                       // r = s*32 + b
            scores[(r & 31) * S_LEN + (r >> 5)] = acc[j];   // [B][S]
        }
    }
}

// --- Kernel 3: per-b softmax over S ---------------------------------------
__global__ void __launch_bounds__(256)
softmax_kernel(const float* __restrict__ scores, float* __restrict__ out) {
    __shared__ float red[256];
    const int b = blockIdx.x;
    const float* sc = scores + b * S_LEN;

    float lmax = -3.402823466e38f;
    for (int i = threadIdx.x; i < S_LEN; i += 256)
        lmax = fmaxf(lmax, sc[i]);
    red[threadIdx.x] = lmax;
    __syncthreads();
    for (int off = 128; off > 0; off >>= 1) {
        if (threadIdx.x < off)
            red[threadIdx.x] = fmaxf(red[threadIdx.x], red[threadIdx.x + off]);
        __syncthreads();
    }
    const float gmax = red[0];
    __syncthreads();

    float lsum = 0.f;
    for (int i = threadIdx.x; i < S_LEN; i += 256)
        lsum += expf(sc[i] - gmax);
    red[threadIdx.x] = lsum;
    __syncthreads();
    for (int off = 128; off > 0; off >>= 1) {
        if (threadIdx.x < off)
            red[threadIdx.x] += red[threadIdx.x + off];
        __syncthreads();
    }
    const float ginv = 1.f / red[0];

    for (int i = threadIdx.x; i < S_LEN; i += 256)
        out[b * S_LEN + i] = expf(sc[i] - gmax) * ginv;   // out is [B,1,S]
}

// ---------------------------------------------------------------------------
extern "C" void kernel_launch(void* const* d_in, const int* in_sizes, int n_in,
                              void* d_out, int out_size, void* d_ws, size_t ws_size,
                              hipStream_t stream) {
    // setup_inputs order: hidden[0], encoder_outputs[1], attn_W[2], attn_b[3], v[4]
    const float* enc = (const float*)d_in[1];
    const float* W   = (const float*)d_in[2];
    const float* v   = (const float*)d_in[4];
    float* out = (float*)d_out;

    float* w2     = (float*)d_ws;         // 1024 floats (4 KB)
    float* scores = w2 + F2;              // 65536 floats (256 KB), [B][S]

    fold_w_kernel    <<<F2 / 256, 256, 0, stream>>>(W, v, w2);
    score_wmma_kernel<<<ROWS / (16 * 8), 256, 0, stream>>>(enc, w2, scores);
    softmax_kernel   <<<B_DIM, 256, 0, stream>>>(scores, out);
}